// DynamicDepthwiseConv_87892210745472
// MI455X (gfx1250) — compile-verified
//
#include <hip/hip_runtime.h>
#include <hip/hip_bf16.h>

// -------- problem constants (match setup_inputs) --------
#define B_   8
#define C_   96
#define H_   256
#define W_   256
#define CH_  12      // c//8 hidden
#define K2_  9       // 3x3 kernel taps
#define PLANE_ (H_ * W_)          // 65536
#define NPLANES_ (B_ * C_)        // 768

typedef __attribute__((ext_vector_type(2))) float v2f;
typedef __attribute__((ext_vector_type(8))) float v8f;

// ============================================================
// Stage 1: global average pool.  pooled[b*C+c] = mean(x[b,c,:,:])
// One 256-thread block per (b,c) plane; float4 streaming loads.
// ============================================================
__global__ __launch_bounds__(256) void pool_kernel(const float* __restrict__ x,
                                                   float* __restrict__ pooled) {
    const int plane = blockIdx.x;                       // b*C + c
    const float4* p = (const float4*)(x + (size_t)plane * PLANE_);
    float s = 0.f;
    for (int it = threadIdx.x; it < PLANE_ / 4; it += 256) {
        float4 v = p[it];
        s += (v.x + v.y) + (v.z + v.w);
    }
    // wave32 reduction
    #pragma unroll
    for (int off = 16; off > 0; off >>= 1)
        s += __shfl_xor(s, off, 32);
    __shared__ float red[8];
    if ((threadIdx.x & 31) == 0) red[threadIdx.x >> 5] = s;
    __syncthreads();
    if (threadIdx.x == 0) {
        float t = 0.f;
        #pragma unroll
        for (int i = 0; i < 8; ++i) t += red[i];
        pooled[plane] = t * (1.0f / (float)PLANE_);
    }
}

// ============================================================
// Stage 2: attention MLP on one wave32 using V_WMMA_F32_16X16X4_F32.
//   hid(8x12) = relu(pooled(8x96) @ w1^T + b1)   -> 24 WMMA k-steps
//   logit(8x9) = hid @ w2^T + b2                 -> 3 WMMA k-steps
//   kw = softmax(logit, axis=1)
// A fragment (16x4 f32, 2 VGPR): lanes 0-15 hold K=k0,k0+1 for M=lane;
//                                lanes 16-31 hold K=k0+2,k0+3 for M=lane-16.
// B fragment (4x16, 2 VGPR): same striping with N across lanes.
// C/D (16x16 f32, 8 VGPR): VGPR r = row M=r (lanes 0-15) / M=r+8 (lanes 16-31).
// Rows/cols beyond the valid 8/12/9 are zero-padded -> exact result.
// ============================================================
__global__ __launch_bounds__(32) void attn_mlp_kernel(
    const float* __restrict__ pooled,   // (8,96)
    const float* __restrict__ w1,       // (12,96)
    const float* __restrict__ b1,       // (12,)
    const float* __restrict__ w2,       // (9,12)
    const float* __restrict__ b2,       // (9,)
    float* __restrict__ kw)             // (8,9)
{
    const int lane = threadIdx.x;
    const int half = lane >> 4;         // 0: K=k0,k0+1 ; 1: K=k0+2,k0+3
    const int l15  = lane & 15;
    __shared__ float sh[16 * 16];

    // ---------- GEMM1: hid = pooled @ w1^T ----------
    v8f c1 = {};
    for (int k0 = 0; k0 < C_; k0 += 4) {
        const int ka = k0 + half * 2;
        v2f a, bm;
        a.x  = (l15 < B_)  ? pooled[l15 * C_ + ka]     : 0.f;
        a.y  = (l15 < B_)  ? pooled[l15 * C_ + ka + 1] : 0.f;
        bm.x = (l15 < CH_) ? w1[l15 * C_ + ka]         : 0.f;  // B[k][n] = w1[n][k]
        bm.y = (l15 < CH_) ? w1[l15 * C_ + ka + 1]     : 0.f;
        c1 = __builtin_amdgcn_wmma_f32_16x16x4_f32(
                 /*neg_a=*/false, a, /*neg_b=*/false, bm,
                 /*c_mod=*/(short)0, c1, /*reuse_a=*/false, /*reuse_b=*/false);
    }
    // spill hid (rows 0..7 live in VGPRs 0..7 at lanes 0..11) through LDS
    if (lane < CH_) {
        #pragma unroll
        for (int r = 0; r < B_; ++r) {
            float h = c1[r] + b1[lane];
            sh[r * CH_ + lane] = h > 0.f ? h : 0.f;
        }
    }
    __syncthreads();

    // ---------- GEMM2: logit = hid @ w2^T ----------
    v8f c2 = {};
    for (int k0 = 0; k0 < CH_; k0 += 4) {
        const int ka = k0 + half * 2;
        v2f a, bm;
        a.x  = (l15 < B_)  ? sh[l15 * CH_ + ka]     : 0.f;
        a.y  = (l15 < B_)  ? sh[l15 * CH_ + ka + 1] : 0.f;
        bm.x = (l15 < K2_) ? w2[l15 * CH_ + ka]     : 0.f;     // B[k][n] = w2[n][k]
        bm.y = (l15 < K2_) ? w2[l15 * CH_ + ka + 1] : 0.f;
        c2 = __builtin_amdgcn_wmma_f32_16x16x4_f32(
                 false, a, false, bm, (short)0, c2, false, false);
    }
    __syncthreads();                    // done reading hid region of sh
    if (lane < K2_) {
        #pragma unroll
        for (int r = 0; r < B_; ++r)
            sh[r * K2_ + lane] = c2[r] + b2[lane];
    }
    __syncthreads();

    // ---------- softmax over the 9 taps, one lane per batch row ----------
    if (lane < B_) {
        float mx = -3.402823466e+38f;
        #pragma unroll
        for (int n = 0; n < K2_; ++n) mx = fmaxf(mx, sh[lane * K2_ + n]);
        float e[K2_];
        float s = 0.f;
        #pragma unroll
        for (int n = 0; n < K2_; ++n) { e[n] = expf(sh[lane * K2_ + n] - mx); s += e[n]; }
        const float inv = 1.f / s;
        #pragma unroll
        for (int n = 0; n < K2_; ++n) kw[lane * K2_ + n] = e[n] * inv;
    }
}

// ============================================================
// Stage 3: dynamic depthwise 3x3 conv (pad=1), bandwidth-bound.
// Each thread produces 4 consecutive outputs in a row via one
// float4 store; per row it needs float4 center + 2 halo scalars.
// ============================================================
__global__ __launch_bounds__(256) void dyndw_kernel(const float* __restrict__ x,
                                                    const float* __restrict__ kwall,
                                                    float* __restrict__ out) {
    const int W4 = W_ / 4;                               // 64
    size_t idx = (size_t)blockIdx.x * 256 + threadIdx.x; // over vec4 outputs
    const size_t total = (size_t)NPLANES_ * H_ * W4;     // 12,582,912
    if (idx >= total) return;

    const int j4 = (int)(idx % W4);
    size_t t1 = idx / W4;
    const int i = (int)(t1 % H_);
    const size_t plane = t1 / H_;                        // b*C + c
    const int b = (int)(plane / C_);
    const float* __restrict__ xp = x + plane * (size_t)PLANE_;

    float k[9];
    #pragma unroll
    for (int p = 0; p < 9; ++p) k[p] = kwall[b * K2_ + p];

    const int j = j4 * 4;
    float acc0 = 0.f, acc1 = 0.f, acc2 = 0.f, acc3 = 0.f;

    #pragma unroll
    for (int di = 0; di < 3; ++di) {
        const int r = i + di - 1;
        float row[6] = {0.f, 0.f, 0.f, 0.f, 0.f, 0.f};
        if (r >= 0 && r < H_) {
            const float* rp = xp + r * W_ + j;
            float4 cv = *(const float4*)rp;              // global_load_b128
            row[1] = cv.x; row[2] = cv.y; row[3] = cv.z; row[4] = cv.w;
            row[0] = (j > 0)        ? rp[-1] : 0.f;      // left halo
            row[5] = (j + 4 < W_)   ? rp[4]  : 0.f;      // right halo
        }
        #pragma unroll
        for (int dj = 0; dj < 3; ++dj) {
            const float kv = k[di * 3 + dj];
            acc0 = fmaf(kv, row[dj + 0], acc0);
            acc1 = fmaf(kv, row[dj + 1], acc1);
            acc2 = fmaf(kv, row[dj + 2], acc2);
            acc3 = fmaf(kv, row[dj + 3], acc3);
        }
    }
    float4 o; o.x = acc0; o.y = acc1; o.z = acc2; o.w = acc3;
    *(float4*)(out + idx * 4) = o;                       // global_store_b128
}

// ============================================================
extern "C" void kernel_launch(void* const* d_in, const int* in_sizes, int n_in,
                              void* d_out, int out_size, void* d_ws, size_t ws_size,
                              hipStream_t stream) {
    const float* x  = (const float*)d_in[0];   // (8,96,256,256)
    const float* w1 = (const float*)d_in[1];   // (12,96)
    const float* b1 = (const float*)d_in[2];   // (12,)
    const float* w2 = (const float*)d_in[3];   // (9,12)
    const float* b2 = (const float*)d_in[4];   // (9,)
    float* out = (float*)d_out;

    float* pooled = (float*)d_ws;              // 768 floats
    float* kw     = pooled + NPLANES_;         // 72 floats

    pool_kernel<<<NPLANES_, 256, 0, stream>>>(x, pooled);
    attn_mlp_kernel<<<1, 32, 0, stream>>>(pooled, w1, b1, w2, b2, kw);

    const size_t total_v4 = (size_t)NPLANES_ * H_ * (W_ / 4);
    const int blocks = (int)((total_v4 + 255) / 256);    // 49152
    dyndw_kernel<<<blocks, 256, 0, stream>>>(x, kw, out);
}